// E3nnSimpleNodeBlock_75471165325410
// MI455X (gfx1250) — compile-verified
//
#include <hip/hip_runtime.h>

typedef float v2f __attribute__((ext_vector_type(2)));
typedef float v4f __attribute__((ext_vector_type(4)));
typedef float v8f __attribute__((ext_vector_type(8)));

#define FEAT_STRIDE 260            // feature row: 16B-aligned, +4 banks/row => conflict-free A loads
#define FEAT_PER_WAVE (16 * FEAT_STRIDE)
#define XS_STRIDE 68               // x row: 16B-aligned, +4 banks/row
#define XS_PER_WAVE (16 * XS_STRIDE)
#define WAVES_PER_WG 4

// path normalizations (double-accurate literals), folded into LDS weights / features
#define A000c 0.044194173824159216f   // sqrt(1/(2*16*16))
#define A110c 0.025515518153991442f   // A000 / sqrt(3)
#define A101c 0.0625f                 // sqrt(3/256)/sqrt(3) = 1/16
#define A112c 0.13975424859373686f    // sqrt(5/256)
#define C10c  0.31622776601683794f    // 1/sqrt(10)
#define C30c  0.18257418583505536f    // 1/sqrt(30)

__launch_bounds__(WAVES_PER_WG * 32)
__attribute__((amdgpu_waves_per_eu(1)))
__global__ void tsq_kernel(const float* __restrict__ feats,
                           const float* __restrict__ msgs,
                           const float* __restrict__ w000,
                           const float* __restrict__ w110,
                           const float* __restrict__ w101,
                           const float* __restrict__ w112,
                           float* __restrict__ out,
                           int N)
{
    extern __shared__ float smem[];
    float* wB      = smem;                          // 4 * 4096: pre-scaled (256x16) weights
    float* featBuf = smem + 4 * 4096;               // WAVES_PER_WG * FEAT_PER_WAVE
    float* xsBuf   = featBuf + WAVES_PER_WG * FEAT_PER_WAVE; // WAVES_PER_WG * XS_PER_WAVE

    const int tid  = threadIdx.x;
    const int wave = tid >> 5;
    const int lane = tid & 31;
    const int lo   = lane & 15;   // node-in-tile (A/D row) == weight column w for B/D
    const int hi   = lane >> 4;   // half-wave selector (K sub-rows / D row offset +8)

    // ---- cooperative: load + scale all four (16,16,16) weight tensors into LDS ----
    {
        const float* src[4] = { w000, w110, w101, w112 };
        const float  scl[4] = { A000c, A110c, A101c, A112c };
        #pragma unroll
        for (int p = 0; p < 4; ++p) {
            for (int i = tid * 4; i < 4096; i += WAVES_PER_WG * 32 * 4) {
                v4f v = *(const v4f*)(src[p] + i);
                v4f o = { v.x * scl[p], v.y * scl[p], v.z * scl[p], v.w * scl[p] };
                *(v4f*)(wB + p * 4096 + i) = o;
            }
        }
    }
    __syncthreads();

    const int tile = blockIdx.x * WAVES_PER_WG + wave;
    const int n0   = tile * 16;
    if (n0 >= N) return;

    float* xs   = xsBuf + wave * XS_PER_WAVE;
    float* feat = featBuf + wave * FEAT_PER_WAVE;

    auto dswait = [&]() { asm volatile("s_wait_dscnt 0" ::: "memory"); };

    // ---- stage x = feats + msgs for the 16-node tile into LDS (half row per lane) ----
    {
        const int r = lane >> 1;           // node row 0..15
        const int c = (lane & 1) * 32;     // column half
        const float* fp = feats + (size_t)(n0 + r) * 64 + c;
        const float* mp = msgs  + (size_t)(n0 + r) * 64 + c;
        float* dst = xs + r * XS_STRIDE + c;
        #pragma unroll
        for (int i = 0; i < 32; i += 4) {
            v4f a = *(const v4f*)(fp + i);
            v4f b = *(const v4f*)(mp + i);
            v4f s = { a.x + b.x, a.y + b.y, a.z + b.z, a.w + b.w };
            *(v4f*)(dst + i) = s;
        }
    }
    dswait();   // per-wave LDS is in-order; fence stage->read for the compiler

    float* frow = feat + lo * FEAT_STRIDE;          // this lane's feature row (node lo)
    const float* xrow = xs + lo * XS_STRIDE;        // this lane's x row (node lo)

    // D[16 nodes x 16 w] += A[16 x 256] * B[256 x 16], 64 steps of V_WMMA_F32_16X16X4_F32
    auto run_gemm = [&](const float* Bl, v8f& acc) {
        const float* aPtr = frow + 2 * hi;          // A frag: row lo, cols 4s+2hi, +1
        const float* bPtr = Bl + hi * 32 + lo;      // B frag: rows 4s+2hi, +1 at col lo
        #pragma unroll 8
        for (int s = 0; s < 64; ++s) {
            v2f a = *(const v2f*)(aPtr + 4 * s);
            v2f b = { bPtr[64 * s], bPtr[64 * s + 16] };
            acc = __builtin_amdgcn_wmma_f32_16x16x4_f32(
                false, a, false, b, (short)0, acc, false, false);
        }
    };

    // acc[r] holds (node n0 + r + 8*hi, w = lo)
    auto store_D = [&](const v8f& acc, int colbase, int cstride) {
        float* base = out + (size_t)(n0 + 8 * hi) * 144 + colbase + cstride * lo;
        #pragma unroll
        for (int r = 0; r < 8; ++r) base[(size_t)r * 144] = acc[r];
    };

    // ---- batch-load the tiny u-side slices into registers (no per-uu LDS round trips) ----
    float su0r[8];                       // x0[u], u = uu + 8*hi
    #pragma unroll
    for (int i = 0; i < 8; i += 4) {
        v4f t = *(const v4f*)(xrow + 8 * hi + i);
        su0r[i] = t.x; su0r[i+1] = t.y; su0r[i+2] = t.z; su0r[i+3] = t.w;
    }
    float su1r[24];                      // x1[u,:], u = uu + 8*hi (3 floats per u)
    #pragma unroll
    for (int i = 0; i < 24; i += 4) {
        v4f t = *(const v4f*)(xrow + 16 + 24 * hi + i);
        su1r[i] = t.x; su1r[i+1] = t.y; su1r[i+2] = t.z; su1r[i+3] = t.w;
    }

    // ======== x0-side phases: register-cache xv0 (16 floats) ========
    float xv0[16];
    #pragma unroll
    for (int i = 0; i < 16; i += 4) {
        v4f t = *(const v4f*)(xrow + i);
        xv0[i] = t.x; xv0[i+1] = t.y; xv0[i+2] = t.z; xv0[i+3] = t.w;
    }

    // path 000 first half: F = x0[u]*x0[v]
    v8f d0 = {};
    #pragma unroll
    for (int uu = 0; uu < 8; ++uu) {
        float s = su0r[uu];
        float* dst = frow + uu * 16 + hi * 128;
        #pragma unroll
        for (int v = 0; v < 16; v += 4) {
            v4f t = { s * xv0[v], s * xv0[v+1], s * xv0[v+2], s * xv0[v+3] };
            *(v4f*)(dst + v) = t;
        }
    }
    dswait();
    run_gemm(wB, d0);                                // keep d0 live

    // path 101: F = x1[u,k]*x0[v] -> out[:, 16+3w+k]
    #pragma unroll
    for (int k = 0; k < 3; ++k) {
        v8f d = {};
        #pragma unroll
        for (int uu = 0; uu < 8; ++uu) {
            float s = su1r[3 * uu + k];
            float* dst = frow + uu * 16 + hi * 128;
            #pragma unroll
            for (int v = 0; v < 16; v += 4) {
                v4f t = { s * xv0[v], s * xv0[v+1], s * xv0[v+2], s * xv0[v+3] };
                *(v4f*)(dst + v) = t;
            }
        }
        dswait();
        run_gemm(wB + 2 * 4096, d);
        store_D(d, 16 + k, 3);
    }

    // ======== x1-side phases: register-cache xv1 (48 floats; xv0 now dead) ========
    float xv1[48];
    #pragma unroll
    for (int i = 0; i < 48; i += 4) {
        v4f t = *(const v4f*)(xrow + 16 + i);
        xv1[i] = t.x; xv1[i+1] = t.y; xv1[i+2] = t.z; xv1[i+3] = t.w;
    }

    // path 110 second half of out0: F = dot(x1[u], x1[v])
    #pragma unroll
    for (int uu = 0; uu < 8; ++uu) {
        float a0 = su1r[3 * uu + 0];
        float a1 = su1r[3 * uu + 1];
        float a2 = su1r[3 * uu + 2];
        float* dst = frow + uu * 16 + hi * 128;
        #pragma unroll
        for (int v = 0; v < 16; ++v)
            dst[v] = a0 * xv1[v*3+0] + a1 * xv1[v*3+1] + a2 * xv1[v*3+2];
    }
    dswait();
    run_gemm(wB + 4096, d0);
    store_D(d0, 0, 1);

    // path 112: F = sum_ij x1[u,i] x1[v,j] C112[i,j,k] -> out[:, 64+5w+k]
    #pragma unroll
    for (int k = 0; k < 5; ++k) {
        v8f d = {};
        #pragma unroll
        for (int uu = 0; uu < 8; ++uu) {
            float b0 = su1r[3 * uu + 0];
            float b1 = su1r[3 * uu + 1];
            float b2 = su1r[3 * uu + 2];
            float* dst = frow + uu * 16 + hi * 128;
            #pragma unroll
            for (int v = 0; v < 16; ++v) {
                float c0 = xv1[v*3+0], c1 = xv1[v*3+1], c2 = xv1[v*3+2];
                float t;
                if      (k == 0) t = (b0*c1 + b1*c0) * C10c;
                else if (k == 1) t = (b0*c2 + b2*c0) * C10c;
                else if (k == 2) t = (b1*c2 + b2*c1) * C10c;
                else if (k == 3) t = (b0*c0 - b1*c1) * C10c;
                else             t = (2.0f*b2*c2 - b0*c0 - b1*c1) * C30c;
                dst[v] = t;
            }
        }
        dswait();
        run_gemm(wB + 3 * 4096, d);
        store_D(d, 64 + k, 5);
    }
}

extern "C" void kernel_launch(void* const* d_in, const int* in_sizes, int n_in,
                              void* d_out, int out_size, void* d_ws, size_t ws_size,
                              hipStream_t stream) {
    const float* feats = (const float*)d_in[0];
    const float* msgs  = (const float*)d_in[1];
    const float* w000  = (const float*)d_in[2];
    const float* w101  = (const float*)d_in[3];
    const float* w110  = (const float*)d_in[4];
    const float* w112  = (const float*)d_in[5];
    float* out = (float*)d_out;

    int N = in_sizes[0] / 64;                 // node count
    int tiles  = (N + 15) / 16;               // 16 nodes per wave
    int blocks = (tiles + WAVES_PER_WG - 1) / WAVES_PER_WG;
    size_t smem = (size_t)(4 * 4096 + WAVES_PER_WG * (FEAT_PER_WAVE + XS_PER_WAVE))
                  * sizeof(float);

    tsq_kernel<<<blocks, WAVES_PER_WG * 32, smem, stream>>>(
        feats, msgs, w000, w110, w101, w112, out, N);
}